// ProteinGINModelSimple_24687472018092
// MI455X (gfx1250) — compile-verified
//
#include <hip/hip_runtime.h>
#include <hip/hip_bf16.h>

#define DD 256   // prot_hidden_dim
#define HH 64    // D/4
#define RR 5     // relation count
#define LN_EPS 1e-5f

typedef __attribute__((ext_vector_type(2))) float v2f;
typedef __attribute__((ext_vector_type(8))) float v8f;

__device__ __forceinline__ void atomAddF32(float* p, float v) {
  // native global_atomic_add_f32 (no CAS loop) on CDNA
  unsafeAtomicAdd(p, v);
}

// ---------------------------------------------------------------------------
// 1) cnt[r][n] = in-degree of node n under relation r (as float)
// ---------------------------------------------------------------------------
__global__ void count_kernel(const int* __restrict__ dst, float* __restrict__ cnt,
                             int E, int N) {
  int e = blockIdx.x * blockDim.x + threadIdx.x;
  int r = blockIdx.y;
  if (e < E) {
    int d = dst[(size_t)r * E + e];
    atomAddF32(&cnt[(size_t)r * N + d], 1.0f);
  }
}

// ---------------------------------------------------------------------------
// 2) cnt -> 1/cnt (0 for zero-degree nodes); implements the where(cnt>0) rule
// ---------------------------------------------------------------------------
__global__ void invert_kernel(float* __restrict__ cnt, int n) {
  int i = blockIdx.x * blockDim.x + threadIdx.x;
  if (i < n) {
    float v = cnt[i];
    cnt[i] = (v > 0.0f) ? (1.0f / v) : 0.0f;
  }
}

// ---------------------------------------------------------------------------
// 3) g[r][d] = sum_e x[src_e][d] * inv[r][dst_e]
//    Block = 256 threads, thread owns column d = tid, walks a chunk of edges.
//    Edge indices are block-uniform -> scalar loads; x row reads are a single
//    fully-coalesced 1KB burst per edge (L2-resident). One atomic per column
//    per block at the end.
// ---------------------------------------------------------------------------
__global__ __launch_bounds__(256) void colsum_kernel(
    const float* __restrict__ x, const int* __restrict__ src,
    const int* __restrict__ dst, const float* __restrict__ inv,
    float* __restrict__ g, int E, int N, int chunk) {
  int r = blockIdx.y;
  int e0 = blockIdx.x * chunk;
  int e1 = min(e0 + chunk, E);
  int tid = threadIdx.x;  // column 0..255
  const int* srcr = src + (size_t)r * E;
  const int* dstr = dst + (size_t)r * E;
  const float* invr = inv + (size_t)r * N;
  float acc = 0.0f;
  for (int e = e0; e < e1; ++e) {
    int s = srcr[e];
    int d = dstr[e];
    float iv = invr[d];
    acc = fmaf(x[(size_t)s * DD + tid], iv, acc);
  }
  atomAddF32(&g[r * DD + tid], acc);
}

// ---------------------------------------------------------------------------
// 4) Relation-attention MLP with V_WMMA_F32_16X16X4_F32.
//    h = (g/N) @ W1 + b1 ; LayerNorm ; ReLU ; score = h@W2 + b2 + bias
//    w = clip(2*sigmoid(score/2), 0.05, 2.0)
//    One wave (32 threads). M padded 5->16 via multiplicative mask (no EXEC
//    divergence around WMMA). A-frag layout per ISA 7.12.2 (32-bit 16x4):
//    lane<16: M=lane, VGPR0=K0,VGPR1=K1 ; lane>=16: M=lane-16, K2/K3.
//    B-frag mirrored: VGPR0 = row K(0|2) striped over 16 lanes, VGPR1 = K(1|3).
// ---------------------------------------------------------------------------
__global__ __launch_bounds__(32) void mlp_kernel(
    const float* __restrict__ g, const float* __restrict__ W1,
    const float* __restrict__ b1, const float* __restrict__ gamma,
    const float* __restrict__ beta, const float* __restrict__ W2,
    const float* __restrict__ b2, const float* __restrict__ attn_bias,
    float* __restrict__ wOut, float invN) {
  __shared__ float hbuf[16 * HH];
  const int lane = threadIdx.x;
  const int half = lane >> 4;   // 0 or 1
  const int idx  = lane & 15;
  const int m    = idx;                       // A row for this lane
  const float amask = (m < RR) ? invN : 0.0f; // fold 1/N scaling + M padding
  const int gm = (m < RR) ? m : 0;            // safe address, masked by amask

  for (int nt = 0; nt < 4; ++nt) {
    const int col = nt * 16 + idx;            // B/D column for this lane
    v8f c = {0.f, 0.f, 0.f, 0.f, 0.f, 0.f, 0.f, 0.f};
    for (int kk = 0; kk < DD / 4; ++kk) {
      const int kbase = kk * 4 + 2 * half;    // K index held by this lane
      v2f a, b;
      a.x = g[gm * DD + kbase    ] * amask;
      a.y = g[gm * DD + kbase + 1] * amask;
      b.x = W1[(size_t)(kbase    ) * HH + col];
      b.y = W1[(size_t)(kbase + 1) * HH + col];
      c = __builtin_amdgcn_wmma_f32_16x16x4_f32(
          /*neg_a=*/false, a, /*neg_b=*/false, b,
          /*c_mod=*/(short)0, c, /*reuse_a=*/false, /*reuse_b=*/false);
    }
    // D layout: VGPR i -> row (i + 8*half), col = 16*nt + idx
#pragma unroll
    for (int i = 0; i < 8; ++i) {
      hbuf[(i + 8 * half) * HH + col] = c[i] + b1[col];
    }
  }
  __syncthreads();

  if (lane < RR) {
    const float* hr = &hbuf[lane * HH];
    float mu = 0.0f;
    for (int j = 0; j < HH; ++j) mu += hr[j];
    mu *= (1.0f / HH);
    float var = 0.0f;
    for (int j = 0; j < HH; ++j) { float t = hr[j] - mu; var = fmaf(t, t, var); }
    var *= (1.0f / HH);
    const float rs = rsqrtf(var + LN_EPS);
    float score = 0.0f;
    for (int j = 0; j < HH; ++j) {
      float hn = (hr[j] - mu) * rs * gamma[j] + beta[j];
      hn = fmaxf(hn, 0.0f);                 // ReLU
      score = fmaf(hn, W2[j], score);
    }
    score += b2[0] + attn_bias[lane];
    float w = 2.0f / (1.0f + expf(-score * 0.5f));  // 2*sigmoid(s/2)
    w = fminf(fmaxf(w, 0.05f), 2.0f);
    wOut[lane] = w;
  }
}

// ---------------------------------------------------------------------------
// 5) out[dst][d] += w[r] * inv[r][dst] * x[src][d]
//    One wave per edge; lane covers 8 columns (2x float4 coalesced reads,
//    8 native f32 atomics into L2-resident out).
// ---------------------------------------------------------------------------
__global__ __launch_bounds__(256) void fuse_kernel(
    const float* __restrict__ x, const int* __restrict__ src,
    const int* __restrict__ dst, const float* __restrict__ inv,
    const float* __restrict__ wv, float* __restrict__ out, int E, int N) {
  const int r = blockIdx.y;
  const int warp = threadIdx.x >> 5;
  const int lane = threadIdx.x & 31;
  const int e = blockIdx.x * 8 + warp;
  if (e >= E) return;
  const int s = src[(size_t)r * E + e];
  const int d = dst[(size_t)r * E + e];
  const float scale = wv[r] * inv[(size_t)r * N + d];
  const float4* xr = (const float4*)(x + (size_t)s * DD) + lane * 2;
  const float4 x0 = xr[0];
  const float4 x1 = xr[1];
  float* o = out + (size_t)d * DD + lane * 8;
  atomAddF32(o + 0, x0.x * scale);
  atomAddF32(o + 1, x0.y * scale);
  atomAddF32(o + 2, x0.z * scale);
  atomAddF32(o + 3, x0.w * scale);
  atomAddF32(o + 4, x1.x * scale);
  atomAddF32(o + 5, x1.y * scale);
  atomAddF32(o + 6, x1.z * scale);
  atomAddF32(o + 7, x1.w * scale);
}

// ---------------------------------------------------------------------------
extern "C" void kernel_launch(void* const* d_in, const int* in_sizes, int n_in,
                              void* d_out, int out_size, void* d_ws, size_t ws_size,
                              hipStream_t stream) {
  const float* x     = (const float*)d_in[0];
  const int*   src   = (const int*)  d_in[1];
  const int*   dst   = (const int*)  d_in[2];
  const float* W1    = (const float*)d_in[3];
  const float* b1    = (const float*)d_in[4];
  const float* gamma = (const float*)d_in[5];
  const float* beta  = (const float*)d_in[6];
  const float* W2    = (const float*)d_in[7];
  const float* b2    = (const float*)d_in[8];
  const float* abias = (const float*)d_in[9];
  float* out = (float*)d_out;

  const int N = in_sizes[0] / DD;  // 50000
  const int E = in_sizes[1] / RR;  // 800000

  // workspace layout (floats): [ cnt/inv : R*N ][ g : R*DD ][ w : R ]
  float* cnt = (float*)d_ws;
  float* g   = cnt + (size_t)RR * N;
  float* wv  = g + RR * DD;
  const size_t ws_floats = (size_t)RR * N + RR * DD + RR;

  hipMemsetAsync(cnt, 0, ws_floats * sizeof(float), stream);
  hipMemsetAsync(out, 0, (size_t)out_size * sizeof(float), stream);

  const dim3 blk(256);
  count_kernel<<<dim3((E + 255) / 256, RR), blk, 0, stream>>>(dst, cnt, E, N);
  invert_kernel<<<dim3((RR * N + 255) / 256), blk, 0, stream>>>(cnt, RR * N);

  const int chunk = 2048;
  colsum_kernel<<<dim3((E + chunk - 1) / chunk, RR), blk, 0, stream>>>(
      x, src, dst, cnt, g, E, N, chunk);

  mlp_kernel<<<1, 32, 0, stream>>>(g, W1, b1, gamma, beta, W2, b2, abias, wv,
                                   1.0f / (float)N);

  fuse_kernel<<<dim3((E + 7) / 8, RR), blk, 0, stream>>>(x, src, dst, cnt, wv,
                                                         out, E, N);
}